// ChebyshevEncoder_80144089743915
// MI455X (gfx1250) — compile-verified
//
#include <hip/hip_runtime.h>

// Chebyshev encoder: out[pt*60 + d*30 + n] = T_n(x[pt].d), pure streaming.
// HBM-bound (252 MB out / 8 MB in -> ~11us at 23.3 TB/s). Full coalescing;
// input staged in via GLOBAL_LOAD_ASYNC_TO_LDS, output streamed out via
// GLOBAL_STORE_ASYNC_FROM_LDS (both on the gfx1250 ASYNCcnt machinery).

#define CHEB_ORDER     30
#define TPB            256
#define FLOATS_PER_PT  (2 * CHEB_ORDER)                 // 60
#define VEC4_PER_THR   (FLOATS_PER_PT / 4)              // 15
#define VEC4_PER_BLK   (TPB * VEC4_PER_THR)             // 3840

typedef __attribute__((ext_vector_type(2))) float v2f;
typedef __attribute__((ext_vector_type(4))) float v4f;
typedef __attribute__((vector_size(8)))  int v2i;       // b64  payload type
typedef __attribute__((vector_size(16))) int v4i;       // b128 payload type

#ifdef __has_builtin
# if __has_builtin(__builtin_amdgcn_global_load_async_to_lds_b64) && \
     __has_builtin(__builtin_amdgcn_s_wait_asynccnt)
#  define CHEB_ASYNC_LD 1
# endif
# if __has_builtin(__builtin_amdgcn_global_store_async_from_lds_b128) && \
     __has_builtin(__builtin_amdgcn_s_wait_asynccnt)
#  define CHEB_ASYNC_ST 1
# endif
#endif
#ifndef CHEB_ASYNC_LD
# define CHEB_ASYNC_LD 0
#endif
#ifndef CHEB_ASYNC_ST
# define CHEB_ASYNC_ST 0
#endif

__global__ __launch_bounds__(TPB)
void cheb_encode_kernel(const v2f* __restrict__ x, v4f* __restrict__ out, int npoints)
{
    __shared__ float s_out[TPB * FLOATS_PER_PT];   // 61,440 B
#if CHEB_ASYNC_LD
    __shared__ v2f  s_in[TPB];                     // 2,048 B
#endif

    const int t  = threadIdx.x;
    const int pt = blockIdx.x * TPB + t;
    const int ptc = pt < npoints ? pt : (npoints - 1);  // clamp OOB lanes

    v2f p;
#if CHEB_ASYNC_LD
    {
        typedef __attribute__((address_space(1))) v2i GV2i;
        typedef __attribute__((address_space(3))) v2i LV2i;
        GV2i* g = (GV2i*)(x + ptc);
        LV2i* l = (LV2i*)&s_in[t];
        __builtin_amdgcn_global_load_async_to_lds_b64(g, l, 0, 1 /*TH=NT*/);
        __builtin_amdgcn_s_wait_asynccnt(0);   // own wave's async lane-loads done
        p = s_in[t];
    }
#else
    p = x[ptc];
#endif

    // Chebyshev recurrence, fully unrolled; consecutive LDS stores merge to b128.
    float* row = &s_out[t * FLOATS_PER_PT];
    #pragma unroll
    for (int d = 0; d < 2; ++d) {
        const float xv = (d == 0) ? p.x : p.y;
        const float x2 = xv + xv;
        float tm1 = 1.0f;
        float tc  = xv;
        row[d * CHEB_ORDER + 0] = 1.0f;
        row[d * CHEB_ORDER + 1] = xv;
        #pragma unroll
        for (int n = 2; n < CHEB_ORDER; ++n) {
            const float tn = __builtin_fmaf(x2, tc, -tm1);
            row[d * CHEB_ORDER + n] = tn;
            tm1 = tc;
            tc  = tn;
        }
    }
    __syncthreads();   // all DS writes complete (dscnt) before streaming out

    // Coalesced flush: block's output region is one contiguous 61,440 B span.
    const size_t base     = (size_t)blockIdx.x * VEC4_PER_BLK;
    const bool full_block = ((size_t)(blockIdx.x + 1) * TPB) <= (size_t)npoints;

#if CHEB_ASYNC_ST
    if (full_block) {
        // Async engine streams LDS -> HBM directly: no VGPR staging, no ds_load.
        typedef __attribute__((address_space(1))) v4i GV4i;
        typedef __attribute__((address_space(3))) v4i LV4i;
        GV4i* gdst = (GV4i*)(out + base);
        LV4i* lsrc = (LV4i*)s_out;
        #pragma unroll
        for (int j = 0; j < VEC4_PER_THR; ++j) {
            __builtin_amdgcn_global_store_async_from_lds_b128(
                gdst + (size_t)j * TPB + t, lsrc + j * TPB + t, 0, 1 /*TH=NT*/);
        }
        __builtin_amdgcn_s_wait_asynccnt(0);   // stores done before LDS teardown
        return;
    }
#endif
    {
        // Fallback / ragged tail: predicated NT b128 stores via VGPRs.
        const v4f* s4 = (const v4f*)s_out;
        const size_t total_v4 = (size_t)npoints * VEC4_PER_THR;
        #pragma unroll
        for (int j = 0; j < VEC4_PER_THR; ++j) {
            const size_t idx = base + (size_t)j * TPB + t;
            if (idx < total_v4) {
                __builtin_nontemporal_store(s4[j * TPB + t], &out[idx]);
            }
        }
    }
}

extern "C" void kernel_launch(void* const* d_in, const int* in_sizes, int n_in,
                              void* d_out, int out_size, void* d_ws, size_t ws_size,
                              hipStream_t stream) {
    (void)n_in; (void)d_ws; (void)ws_size; (void)out_size;
    const v2f* x = (const v2f*)d_in[0];
    v4f* out = (v4f*)d_out;
    const int npoints = in_sizes[0] / 2;              // 16*65536 = 1,048,576
    const int blocks  = (npoints + TPB - 1) / TPB;    // 4096
    cheb_encode_kernel<<<blocks, TPB, 0, stream>>>(x, out, npoints);
}